// MambaMomentum_67723044323810
// MI455X (gfx1250) — compile-verified
//
#include <hip/hip_runtime.h>
#include <math.h>

typedef __attribute__((ext_vector_type(2))) float v2f;
typedef __attribute__((ext_vector_type(8))) float v8f;
typedef __attribute__((ext_vector_type(4))) int   v4i;

#define D_MODEL   1024
#define D_STATE   16
#define D_CONV    4
#define D_INNER   2048
#define DT_RANK   64
#define BATCH     2
#define SEQLEN    1024
#define NROWS     (BATCH * SEQLEN)          // 2048
#define XPROJ_OUT (DT_RANK + 2 * D_STATE)   // 96
#define BETA      0.6f
#define EPS       1e-5f

// GEMM tiling: 256 threads = 8 waves = 4(M) x 2(N) tiles of 16x16 -> 64x32 block
#define KC        32                         // K chunk staged in LDS
#define LDS_PITCH 36                         // 16B-aligned rows, bank-conflict-free

__device__ __forceinline__ float silu_f(float v) {
  return v / (1.0f + __expf(-v));
}
__device__ __forceinline__ float softplus_f(float v) {
  return (v > 20.0f) ? v : log1pf(__expf(v));
}

// ---------------------------------------------------------------------------
// CDNA5 async global->LDS copy (16 bytes / lane), tracked by ASYNCcnt.
// Builtin signature (from hipcc diagnostic): (v4i AS1*, v4i AS3*, imm, imm).
// ---------------------------------------------------------------------------
typedef __attribute__((address_space(1))) v4i as1_v4i;
typedef __attribute__((address_space(3))) v4i as3_v4i;

__device__ __forceinline__ void async_copy16(const float* g, float* s) {
#if __has_builtin(__builtin_amdgcn_global_load_async_to_lds_b128)
  __builtin_amdgcn_global_load_async_to_lds_b128((as1_v4i*)g, (as3_v4i*)s, 0, 0);
#else
  // VGLOBAL encoding: VDST = LDS address, VADDR = 64-bit global address
  asm volatile("global_load_async_to_lds_b128 %0, %1, off"
               :: "v"((unsigned int)(unsigned long long)s),
                  "v"((unsigned long long)g)
               : "memory");
#endif
}

__device__ __forceinline__ void wait_async0() {
#if __has_builtin(__builtin_amdgcn_s_wait_asynccnt)
  __builtin_amdgcn_s_wait_asynccnt(0);
#else
  asm volatile("s_wait_asynccnt 0x0" ::: "memory");
#endif
}

// ---------------------------------------------------------------------------
// Block GEMM core: C_tile(64x32) = A(64xK) * W^T, W is (N,K) row-major.
// Double-buffered LDS staging via async copies; fp32 WMMA 16x16x4 per ISA
// 7.12.2 layout (A lane r: k+2*half..+1 of row m0+r; B == W rows identically).
// ---------------------------------------------------------------------------
__device__ __forceinline__ v8f gemm_block_compute(const float* __restrict__ A, int lda,
                                                  const float* __restrict__ W, int ldw,
                                                  int K, int mBlk, int nBlk) {
  __shared__ __align__(16) float sA[2][64 * LDS_PITCH];
  __shared__ __align__(16) float sW[2][32 * LDS_PITCH];

  const int tid  = threadIdx.x;
  const int wave = tid >> 5;
  const int lane = tid & 31;
  const int half = lane >> 4;
  const int r    = lane & 15;
  const int wm   = wave & 3;    // 0..3 -> M sub-tile
  const int wn   = wave >> 2;   // 0..1 -> N sub-tile

  auto issue = [&](int chunk, int buf) {
    const int kc = chunk * KC;
    // A chunk: 64 rows x 8 float4 = 512 units, 2 per thread (coalesced)
    int u = tid;
    int row = u >> 3, c4 = (u & 7) << 2;
    async_copy16(A + (size_t)(mBlk + row) * lda + kc + c4,
                 &sA[buf][row * LDS_PITCH + c4]);
    u = tid + 256; row = u >> 3; c4 = (u & 7) << 2;
    async_copy16(A + (size_t)(mBlk + row) * lda + kc + c4,
                 &sA[buf][row * LDS_PITCH + c4]);
    // W chunk: 32 rows x 8 float4 = 256 units, 1 per thread
    row = tid >> 3; c4 = (tid & 7) << 2;
    async_copy16(W + (size_t)(nBlk + row) * ldw + kc + c4,
                 &sW[buf][row * LDS_PITCH + c4]);
  };

  v8f acc = {0.f, 0.f, 0.f, 0.f, 0.f, 0.f, 0.f, 0.f};
  const int nChunks = K / KC;

  issue(0, 0);
  wait_async0();
  __syncthreads();

  int cur = 0;
  for (int c = 0; c < nChunks; ++c) {
    if (c + 1 < nChunks) issue(c + 1, cur ^ 1);

    const float* aP = &sA[cur][(wm * 16 + r) * LDS_PITCH + half * 2];
    const float* wP = &sW[cur][(wn * 16 + r) * LDS_PITCH + half * 2];
#pragma unroll
    for (int k = 0; k < KC; k += 4) {
      v2f a = *(const v2f*)(aP + k);   // ds_load_b64, conflict-free (pitch 36)
      v2f b = *(const v2f*)(wP + k);
      acc = __builtin_amdgcn_wmma_f32_16x16x4_f32(
          /*neg_a=*/false, a, /*neg_b=*/false, b,
          /*c_mod=*/(short)0, acc, /*reuse_a=*/false, /*reuse_b=*/false);
    }

    wait_async0();      // next buffer's async writes landed
    __syncthreads();    // everyone done reading cur / writing next
    cur ^= 1;
  }
  return acc;
}

// ---- GEMM kernels (grid: x = N/32, y = M/64; all dims divide exactly) ------
__global__ void gemm_store_kernel(const float* __restrict__ A, int lda,
                                  const float* __restrict__ W, int ldw,
                                  float* __restrict__ C, int ldc, int K) {
  const int mBlk = blockIdx.y * 64, nBlk = blockIdx.x * 32;
  v8f acc = gemm_block_compute(A, lda, W, ldw, K, mBlk, nBlk);
  const int wave = threadIdx.x >> 5, lane = threadIdx.x & 31;
  const int half = lane >> 4, r = lane & 15;
  const int m0 = mBlk + (wave & 3) * 16, n0 = nBlk + (wave >> 2) * 16;
  for (int i = 0; i < 8; ++i)
    C[(size_t)(m0 + half * 8 + i) * ldc + n0 + r] = acc[i];
}

__global__ void gemm_bias_softplus_kernel(const float* __restrict__ A, int lda,
                                          const float* __restrict__ W, int ldw,
                                          const float* __restrict__ bias,
                                          float* __restrict__ C, int ldc, int K) {
  const int mBlk = blockIdx.y * 64, nBlk = blockIdx.x * 32;
  v8f acc = gemm_block_compute(A, lda, W, ldw, K, mBlk, nBlk);
  const int wave = threadIdx.x >> 5, lane = threadIdx.x & 31;
  const int half = lane >> 4, r = lane & 15;
  const int m0 = mBlk + (wave & 3) * 16, n0 = nBlk + (wave >> 2) * 16;
  const float b = bias[n0 + r];
  for (int i = 0; i < 8; ++i)
    C[(size_t)(m0 + half * 8 + i) * ldc + n0 + r] = softplus_f(acc[i] + b);
}

__global__ void gemm_residual_kernel(const float* __restrict__ A, int lda,
                                     const float* __restrict__ W, int ldw,
                                     const float* __restrict__ R,
                                     float* __restrict__ C, int ldc, int K) {
  const int mBlk = blockIdx.y * 64, nBlk = blockIdx.x * 32;
  v8f acc = gemm_block_compute(A, lda, W, ldw, K, mBlk, nBlk);
  const int wave = threadIdx.x >> 5, lane = threadIdx.x & 31;
  const int half = lane >> 4, r = lane & 15;
  const int m0 = mBlk + (wave & 3) * 16, n0 = nBlk + (wave >> 2) * 16;
  for (int i = 0; i < 8; ++i) {
    const size_t idx = (size_t)(m0 + half * 8 + i) * ldc + n0 + r;
    C[idx] = acc[i] + R[idx];
  }
}

// ---- RMSNorm: one block per row, 256 threads -------------------------------
__global__ void rmsnorm_kernel(const float* __restrict__ x,
                               const float* __restrict__ w,
                               float* __restrict__ out) {
  __shared__ float red[8];
  const size_t row = blockIdx.x;
  const float* xr = x + row * D_MODEL;
  float ss = 0.f;
  for (int i = threadIdx.x; i < D_MODEL; i += 256) {
    float v = xr[i];
    ss += v * v;
  }
  for (int m = 16; m >= 1; m >>= 1) ss += __shfl_xor(ss, m, 32);
  if ((threadIdx.x & 31) == 0) red[threadIdx.x >> 5] = ss;
  __syncthreads();
  float tot = 0.f;
  for (int i = 0; i < 8; ++i) tot += red[i];
  const float scale = rsqrtf(tot * (1.0f / D_MODEL) + EPS);
  for (int i = threadIdx.x; i < D_MODEL; i += 256)
    out[row * D_MODEL + i] = xr[i] * scale * w[i];
}

// ---- Depthwise causal conv (4 taps) + bias + SiLU --------------------------
__global__ void conv_silu_kernel(const float* __restrict__ xz,   // (NROWS, 2*D_INNER)
                                 const float* __restrict__ cw,   // (D_INNER, D_CONV)
                                 const float* __restrict__ cb,   // (D_INNER)
                                 float* __restrict__ xb) {       // (NROWS, D_INNER)
  const size_t idx = (size_t)blockIdx.x * blockDim.x + threadIdx.x;
  if (idx >= (size_t)NROWS * D_INNER) return;
  const int e = (int)(idx % D_INNER);
  const int l = (int)((idx / D_INNER) % SEQLEN);
  const int b = (int)(idx / ((size_t)D_INNER * SEQLEN));
  float acc = cb[e];
  for (int k = 0; k < D_CONV; ++k) {
    const int ls = l + k - (D_CONV - 1);
    if (ls >= 0)
      acc += xz[((size_t)b * SEQLEN + ls) * (2 * D_INNER) + e] * cw[e * D_CONV + k];
  }
  xb[idx] = silu_f(acc);
}

// ---- Selective scan with momentum ------------------------------------------
// One channel (b,e) per 16-lane half-wave; lane n = state index. y overwrites
// delta in place (read-before-write within each step).
__global__ void scan_kernel(float* __restrict__ dy,            // (NROWS, D_INNER) delta in / y out
                            const float* __restrict__ xb,      // (NROWS, D_INNER)
                            const float* __restrict__ dbc,     // (NROWS, 96)
                            const float* __restrict__ A_log) { // (D_INNER, 16)
  const int n      = threadIdx.x & 15;
  const int halfId = threadIdx.x >> 4;            // 0..15 within block
  const int ch     = blockIdx.x * 16 + halfId;    // 0..BATCH*D_INNER-1
  const int b = ch / D_INNER;
  const int e = ch % D_INNER;
  const float An = -__expf(A_log[e * D_STATE + n]);
  float h = 0.f, v = 0.f;
  for (int l = 0; l < SEQLEN; ++l) {
    const size_t row = (size_t)b * SEQLEN + l;
    const float d  = dy[row * D_INNER + e];
    const float xv = xb[row * D_INNER + e];
    const float Bn = dbc[row * XPROJ_OUT + DT_RANK + n];
    const float Cn = dbc[row * XPROJ_OUT + DT_RANK + D_STATE + n];
    const float dA = __expf(d * An);
    v = BETA * v + d * Bn * xv;                   // ALPHA == 1
    h = dA * h + v;
    float yp = h * Cn;
    yp += __shfl_xor(yp, 1, 32);                  // masks <16 stay in half-wave
    yp += __shfl_xor(yp, 2, 32);
    yp += __shfl_xor(yp, 4, 32);
    yp += __shfl_xor(yp, 8, 32);
    if (n == 0) dy[row * D_INNER + e] = yp;
  }
}

// ---- Gate: y = (y + Dp*xb) * silu(z) ---------------------------------------
__global__ void gate_kernel(float* __restrict__ y,
                            const float* __restrict__ xb,
                            const float* __restrict__ xz,
                            const float* __restrict__ Dp) {
  const size_t idx = (size_t)blockIdx.x * blockDim.x + threadIdx.x;
  if (idx >= (size_t)NROWS * D_INNER) return;
  const int e = (int)(idx % D_INNER);
  const size_t row = idx / D_INNER;
  const float z = xz[row * (2 * D_INNER) + D_INNER + e];
  y[idx] = (y[idx] + Dp[e] * xb[idx]) * silu_f(z);
}

// ---------------------------------------------------------------------------
extern "C" void kernel_launch(void* const* d_in, const int* in_sizes, int n_in,
                              void* d_out, int out_size, void* d_ws, size_t ws_size,
                              hipStream_t stream) {
  (void)in_sizes; (void)n_in; (void)out_size; (void)ws_size;

  const float* x_in  = (const float*)d_in[0];
  const float* normw = (const float*)d_in[1];
  const float* inw   = (const float*)d_in[2];
  const float* convw = (const float*)d_in[3];
  const float* convb = (const float*)d_in[4];
  const float* xpw   = (const float*)d_in[5];
  const float* dtw   = (const float*)d_in[6];
  const float* dtb   = (const float*)d_in[7];
  const float* alog  = (const float*)d_in[8];
  const float* dp    = (const float*)d_in[9];
  const float* outw  = (const float*)d_in[10];

  float* XN   = (float*)d_ws;                               // 2048*1024
  float* XZ   = XN  + (size_t)NROWS * D_MODEL;              // 2048*4096
  float* XB   = XZ  + (size_t)NROWS * 2 * D_INNER;          // 2048*2048
  float* DBC  = XB  + (size_t)NROWS * D_INNER;              // 2048*96
  float* DLT  = DBC + (size_t)NROWS * XPROJ_OUT;            // 2048*2048 (delta -> y)
  float* XMID = DLT + (size_t)NROWS * D_INNER;              // 2048*1024

  const int elemBlocks = (int)(((size_t)NROWS * D_INNER + 255) / 256);

  for (int layer = 0; layer < 2; ++layer) {
    const float* nw_l = normw + (size_t)layer * D_MODEL;
    const float* iw_l = inw   + (size_t)layer * 2 * D_INNER * D_MODEL;
    const float* cw_l = convw + (size_t)layer * D_INNER * D_CONV;
    const float* cb_l = convb + (size_t)layer * D_INNER;
    const float* xp_l = xpw   + (size_t)layer * XPROJ_OUT * D_INNER;
    const float* dw_l = dtw   + (size_t)layer * D_INNER * DT_RANK;
    const float* db_l = dtb   + (size_t)layer * D_INNER;
    const float* al_l = alog  + (size_t)layer * D_INNER * D_STATE;
    const float* dp_l = dp    + (size_t)layer * D_INNER;
    const float* ow_l = outw  + (size_t)layer * D_MODEL * D_INNER;

    const float* xin  = (layer == 0) ? x_in : XMID;
    float*       xout = (layer == 1) ? (float*)d_out : XMID;

    // 1) RMSNorm
    rmsnorm_kernel<<<NROWS, 256, 0, stream>>>(xin, nw_l, XN);

    // 2) in_proj: (2048x1024)x(1024->4096) -> XZ
    gemm_store_kernel<<<dim3(2 * D_INNER / 32, NROWS / 64), 256, 0, stream>>>(
        XN, D_MODEL, iw_l, D_MODEL, XZ, 2 * D_INNER, D_MODEL);

    // 3) depthwise conv + SiLU -> XB
    conv_silu_kernel<<<elemBlocks, 256, 0, stream>>>(XZ, cw_l, cb_l, XB);

    // 4) x_proj: (2048x2048)x(2048->96) -> DBC
    gemm_store_kernel<<<dim3(XPROJ_OUT / 32, NROWS / 64), 256, 0, stream>>>(
        XB, D_INNER, xp_l, D_INNER, DBC, XPROJ_OUT, D_INNER);

    // 5) dt proj + bias + softplus: (2048x64)x(64->2048) -> DLT
    gemm_bias_softplus_kernel<<<dim3(D_INNER / 32, NROWS / 64), 256, 0, stream>>>(
        DBC, XPROJ_OUT, dw_l, DT_RANK, db_l, DLT, D_INNER, DT_RANK);

    // 6) selective scan (delta -> y in place)
    scan_kernel<<<(BATCH * D_INNER) / 16, 256, 0, stream>>>(DLT, XB, DBC, al_l);

    // 7) gate
    gate_kernel<<<elemBlocks, 256, 0, stream>>>(DLT, XB, XZ, dp_l);

    // 8) out_proj + residual: (2048x2048)x(2048->1024) + xin -> xout
    gemm_residual_kernel<<<dim3(D_MODEL / 32, NROWS / 64), 256, 0, stream>>>(
        DLT, D_INNER, ow_l, D_INNER, xin, xout, D_MODEL, D_INNER);
  }
}